// ValueNet_4054449127653
// MI455X (gfx1250) — compile-verified
//
#include <hip/hip_runtime.h>
#include <hip/hip_bf16.h>

typedef __attribute__((ext_vector_type(2))) float v2f;
typedef __attribute__((ext_vector_type(8))) float v8f;

#define Nn 50000
#define Ee 600000
#define Hh 128
#define Gg 64
#define LDS_STRIDE 260   // 256 cols + 4 pad -> conflict-free strided fragment reads

// CDNA5 hardware fp32 atomic add, device scope, no return (STOREcnt path, RMW in L2)
__device__ __forceinline__ void atomic_fadd_dev(float* p, float v) {
  asm volatile("global_atomic_add_f32 %0, %1, off scope:SCOPE_DEV"
               : : "v"(p), "v"(v) : "memory");
}

// ---------------- utility kernels ----------------

__global__ void fill0_f4(float4* __restrict__ p, long n4) {
  long i = (long)blockIdx.x * blockDim.x + threadIdx.x;
  long stride = (long)gridDim.x * blockDim.x;
  float4 z = {0.f, 0.f, 0.f, 0.f};
  for (; i < n4; i += stride) p[i] = z;
}

__global__ void count_edges(const int* __restrict__ dst, int E, float* __restrict__ cnt) {
  int e = blockIdx.x * blockDim.x + threadIdx.x;
  if (e < E) atomic_fadd_dev(&cnt[dst[e]], 1.0f);
}

// one wave (32 lanes) per edge: gather 128 floats of x[src], scatter-add into agg[dst]
__global__ void scatter_add(const float* __restrict__ x, const int* __restrict__ src,
                            const int* __restrict__ dst, int E, float* __restrict__ agg) {
  int e = (blockIdx.x * blockDim.x + threadIdx.x) >> 5;
  int lane = threadIdx.x & 31;
  if (e >= E) return;
  int s = src[e];
  int d = dst[e];
  float4 v = ((const float4*)(x + (size_t)s * Hh))[lane];
  float* a = agg + (size_t)d * Hh + lane * 4;
  atomic_fadd_dev(a + 0, v.x);
  atomic_fadd_dev(a + 1, v.y);
  atomic_fadd_dev(a + 2, v.z);
  atomic_fadd_dev(a + 3, v.w);
}

// one wave per node: g[batch[i]] += h[i]
__global__ void pool_add(const float* __restrict__ h, const int* __restrict__ batch,
                         int n, float* __restrict__ g) {
  int node = (blockIdx.x * blockDim.x + threadIdx.x) >> 5;
  int lane = threadIdx.x & 31;
  if (node >= n) return;
  int b = batch[node];
  float4 v = ((const float4*)(h + (size_t)node * Hh))[lane];
  float* gp = g + (size_t)b * Hh + lane * 4;
  atomic_fadd_dev(gp + 0, v.x);
  atomic_fadd_dev(gp + 1, v.y);
  atomic_fadd_dev(gp + 2, v.z);
  atomic_fadd_dev(gp + 3, v.w);
}

// ------- fused SAGE GEMM: out = norm?( (agg/max(cnt,1)) @ Wl + in @ Wr + bl ) -------
// One wave per 32-row strip (M-blocked x2 so each B fragment feeds 2 WMMAs).
// A = [agg_scaled | in] (32 x 256) staged in padded LDS; K=256 via 64 f32 16x16x4 steps.
// L2 row-normalize fused in the epilogue (4 x shfl_xor within 16-lane halves).
__global__ void __launch_bounds__(32)
sage_gemm_wmma(const float* __restrict__ in, const float* __restrict__ agg,
               const float* __restrict__ cnt, const float* __restrict__ Wl,
               const float* __restrict__ Wr, const float* __restrict__ bl,
               float* __restrict__ out, int nrows, int donorm) {
  extern __shared__ float lds[];     // 32 * LDS_STRIDE floats
  const int lane = threadIdx.x & 31;
  const int row0 = blockIdx.x * 32;
  if (row0 >= nrows) return;
  float* A = lds;

  // Stage 32 rows x 256 cols: [0,128) = agg/max(cnt,1), [128,256) = in; tail rows = 0
  for (int c4 = lane; c4 < 32 * 64; c4 += 32) {
    int r = c4 >> 6;
    int c = c4 & 63;
    int grow = row0 + r;
    float4 v = {0.f, 0.f, 0.f, 0.f};
    if (grow < nrows) {
      if (c < 32) {
        v = ((const float4*)(agg + (size_t)grow * Hh))[c];
        float inv = 1.0f / fmaxf(cnt[grow], 1.0f);
        v.x *= inv; v.y *= inv; v.z *= inv; v.w *= inv;
      } else {
        v = ((const float4*)(in + (size_t)grow * Hh))[c - 32];
      }
    }
    *(float4*)(A + r * LDS_STRIDE + c * 4) = v;   // 16B aligned (1040B row stride)
  }
  asm volatile("s_wait_dscnt 0x0" ::: "memory");  // own-wave staging stores complete

  v8f acc[2][8];
  #pragma unroll
  for (int s = 0; s < 2; ++s)
    #pragma unroll
    for (int j = 0; j < 8; ++j)
      acc[s][j] = (v8f){0.f,0.f,0.f,0.f,0.f,0.f,0.f,0.f};

  const int arow = lane & 15;          // A: M = lane%16 (+16 for second strip)
  const int koff = (lane >> 4) << 1;   // K offset 0 or 2 within the 4-block
  const int bn   = lane & 15;          // B: N = lane%16

  #pragma unroll 1
  for (int half = 0; half < 2; ++half) {
    const float* W  = half ? Wr : Wl;          // row-major H x H
    const float* Ah = A + half * 128;
    #pragma unroll 1
    for (int kk = 0; kk < 128; kk += 4) {
      v2f a0 = *(const v2f*)(Ah + arow * LDS_STRIDE + kk + koff);
      v2f a1 = *(const v2f*)(Ah + (arow + 16) * LDS_STRIDE + kk + koff);
      const float* wp0 = W + (kk + koff) * Hh + bn;   // W[k+koff][n]
      const float* wp1 = wp0 + Hh;                    // W[k+koff+1][n]
      #pragma unroll
      for (int j = 0; j < 8; ++j) {
        v2f b;
        b.x = wp0[j * 16];
        b.y = wp1[j * 16];
        acc[0][j] = __builtin_amdgcn_wmma_f32_16x16x4_f32(
            false, a0, false, b, (short)0, acc[0][j], false, false);
        acc[1][j] = __builtin_amdgcn_wmma_f32_16x16x4_f32(
            false, a1, false, b, (short)0, acc[1][j], false, false);
      }
    }
  }

  // Epilogue: +bias, optional L2 row-normalize, store.
  // C/D layout: VGPR r -> M=r (lanes 0-15) / M=8+r (lanes 16-31), N = lane%16.
  const int cc = lane & 15;
  #pragma unroll
  for (int s = 0; s < 2; ++s)
    #pragma unroll
    for (int j = 0; j < 8; ++j) {
      float bias = bl[j * 16 + cc];
      #pragma unroll
      for (int r = 0; r < 8; ++r) acc[s][j][r] += bias;
    }

  if (donorm) {
    #pragma unroll
    for (int s = 0; s < 2; ++s)
      #pragma unroll
      for (int r = 0; r < 8; ++r) {
        float ss = 0.f;
        #pragma unroll
        for (int j = 0; j < 8; ++j) ss += acc[s][j][r] * acc[s][j][r];
        ss += __shfl_xor(ss, 1);   // reductions stay inside each 16-lane half,
        ss += __shfl_xor(ss, 2);   // which is exactly one output row
        ss += __shfl_xor(ss, 4);
        ss += __shfl_xor(ss, 8);
        float sc = 1.0f / fmaxf(sqrtf(ss), 1e-12f);
        #pragma unroll
        for (int j = 0; j < 8; ++j) acc[s][j][r] *= sc;
      }
  }

  const int rbh = (lane >> 4) << 3;    // +8 for upper-half lanes
  #pragma unroll
  for (int s = 0; s < 2; ++s) {
    int rb = row0 + s * 16 + rbh;
    #pragma unroll
    for (int j = 0; j < 8; ++j) {
      float* op = out + (size_t)rb * Hh + j * 16 + cc;
      #pragma unroll
      for (int r = 0; r < 8; ++r)
        if (rb + r < nrows) op[(size_t)r * Hh] = acc[s][j][r];
    }
  }
}

// ---------------- fused MLP head: relu(relu(g@W0+b0)@W1+b1)@hw + hb ----------------
__global__ void __launch_bounds__(256)
mlp_head(const float* __restrict__ g, const float* __restrict__ W0, const float* __restrict__ b0,
         const float* __restrict__ W1, const float* __restrict__ b1,
         const float* __restrict__ hw, const float* __restrict__ hb, float* __restrict__ out) {
  __shared__ float sbuf[Gg * Hh];   // 32 KB
  const int tid = threadIdx.x;

  for (int o = tid; o < Gg * Hh; o += 256) {
    int i = o >> 7, c = o & 127;
    float acc = b0[c];
    const float* gi = g + i * Hh;
    #pragma unroll 4
    for (int k = 0; k < Hh; ++k) acc += gi[k] * W0[k * Hh + c];
    sbuf[o] = fmaxf(acc, 0.0f);
  }
  __syncthreads();

  float r[32];
  int idx = 0;
  for (int o = tid; o < Gg * Hh; o += 256, ++idx) {
    int i = o >> 7, c = o & 127;
    float acc = b1[c];
    const float* si = sbuf + i * Hh;
    #pragma unroll 4
    for (int k = 0; k < Hh; ++k) acc += si[k] * W1[k * Hh + c];
    r[idx] = fmaxf(acc, 0.0f);
  }
  __syncthreads();
  idx = 0;
  for (int o = tid; o < Gg * Hh; o += 256, ++idx) sbuf[o] = r[idx];
  __syncthreads();

  if (tid < Gg) {
    float acc = hb[0];
    const float* si = sbuf + tid * Hh;
    #pragma unroll 4
    for (int k = 0; k < Hh; ++k) acc += si[k] * hw[k];
    out[tid] = acc;
  }
}

// ---------------- host orchestration ----------------

static void run_sage(const float* in, const int* edge, const float* Wl, const float* bl,
                     const float* Wr, const float* cnt, int donorm, float* agg, float* out,
                     hipStream_t stream) {
  fill0_f4<<<2048, 256, 0, stream>>>((float4*)agg, (long)Nn * Hh / 4);
  scatter_add<<<(Ee * 32) / 256, 256, 0, stream>>>(in, edge, edge + Ee, Ee, agg);
  int nblocks = (Nn + 31) / 32;                         // 1563 strips of 32 rows
  size_t ldsb = 32 * LDS_STRIDE * sizeof(float);        // 33,280 B / workgroup (1 wave)
  sage_gemm_wmma<<<nblocks, 32, ldsb, stream>>>(in, agg, cnt, Wl, Wr, bl, out, Nn, donorm);
}

extern "C" void kernel_launch(void* const* d_in, const int* in_sizes, int n_in,
                              void* d_out, int out_size, void* d_ws, size_t ws_size,
                              hipStream_t stream) {
  const float* x     = (const float*)d_in[0];
  const int*   eic   = (const int*)d_in[1];
  const int*   eid   = (const int*)d_in[2];
  const int*   eit   = (const int*)d_in[3];
  const int*   batch = (const int*)d_in[4];
  auto F = [&](int i) { return (const float*)d_in[i]; };
  const float *c1Wl=F(5),  *c1bl=F(6),  *c1Wr=F(7);
  const float *c2Wl=F(8),  *c2bl=F(9),  *c2Wr=F(10);
  const float *c3Wl=F(11), *c3bl=F(12), *c3Wr=F(13);
  const float *c4Wl=F(14), *c4bl=F(15), *c4Wr=F(16);
  const float *c5Wl=F(17), *c5bl=F(18), *c5Wr=F(19);
  const float *l0W=F(20), *l0b=F(21), *l1W=F(22), *l1b=F(23), *hW=F(24), *hb=F(25);

  // workspace layout (floats)
  float* hA   = (float*)d_ws;
  float* hB   = hA + (size_t)Nn * Hh;
  float* agg  = hB + (size_t)Nn * Hh;
  float* cntC = agg + (size_t)Nn * Hh;
  float* cntD = cntC + Nn;
  float* cntT = cntD + Nn;
  float* g    = cntT + Nn;               // Gg*Hh

  // zero counts + pooled buffer (contiguous region)
  fill0_f4<<<512, 256, 0, stream>>>((float4*)cntC, (long)(3 * Nn + Gg * Hh) / 4);
  count_edges<<<(Ee + 255) / 256, 256, 0, stream>>>(eic + Ee, Ee, cntC);
  count_edges<<<(Ee + 255) / 256, 256, 0, stream>>>(eid + Ee, Ee, cntD);
  count_edges<<<(Ee + 255) / 256, 256, 0, stream>>>(eit + Ee, Ee, cntT);

  run_sage(x,  eid, c1Wl, c1bl, c1Wr, cntD, 1, agg, hA, stream);
  run_sage(hA, eic, c2Wl, c2bl, c2Wr, cntC, 1, agg, hB, stream);
  run_sage(hB, eic, c2Wl, c2bl, c2Wr, cntC, 1, agg, hA, stream);
  run_sage(hA, eit, c3Wl, c3bl, c3Wr, cntT, 0, agg, hB, stream);
  run_sage(hB, eid, c4Wl, c4bl, c4Wr, cntD, 1, agg, hA, stream);
  run_sage(hA, eic, c5Wl, c5bl, c5Wr, cntC, 1, agg, hB, stream);
  run_sage(hB, eic, c5Wl, c5bl, c5Wr, cntC, 1, agg, hA, stream);

  pool_add<<<(Nn * 32 + 255) / 256, 256, 0, stream>>>(hA, batch, Nn, g);
  mlp_head<<<1, 256, 0, stream>>>(g, l0W, l0b, l1W, l1b, hW, hb, (float*)d_out);
}